// EdgeEmbedding_86242943304324
// MI455X (gfx1250) — compile-verified
//
#include <hip/hip_runtime.h>

#define BB 8
#define LL 256
#define AAT 15            // atoms per block (real)
#define KK 9
#define ES 16
#define NB (BB*LL)        // 2048 blocks total
#define NEH (NB*KK)       // 18432 edges per half
#define NE (2*NEH)        // 36864 edges

#define N_BLOCKID (BB*LL*AAT)            // 30720
#define OFS_BATCH N_BLOCKID              // 30720
#define OFS_EDGES (N_BLOCKID + NB)       // 32768
#define OFS_ATTR  (OFS_EDGES + 2*NE)     // 106496

// workspace layout in floats
#define WS_A   0
#define WS_B   (NB*32*2)      // 131072
#define WS_X2  (2*NB*32*2)    // 262144
#define WS_SEG (WS_X2 + NB*16) // 294912 (64 u32 words: seg bitmap per batch)

typedef __attribute__((ext_vector_type(2))) float v2f;
typedef __attribute__((ext_vector_type(8))) float v8f;

// ---- raw VALU min ops (skip fminf's NaN canonicalization; inputs always finite) ----
__device__ __forceinline__ float vminf(float a, float b) {
    float r;
    asm("v_min_num_f32_e32 %0, %1, %2" : "=v"(r) : "v"(a), "v"(b));
    return r;
}
__device__ __forceinline__ float vmin3f(float a, float b, float c) {
    float r;
    asm("v_min3_num_f32 %0, %1, %2, %3" : "=v"(r) : "v"(a), "v"(b), "v"(c));
    return r;
}
// fused DPP-min: one instruction per butterfly step
#define DPP_MIN(res, src, DPPCTRL)                                                   \
    asm("v_min_num_f32_dpp %0, %1, %1 " DPPCTRL " row_mask:0xf bank_mask:0xf bound_ctrl:1" \
        : "=v"(res) : "v"(src))

__device__ __forceinline__ float wave32_min(float m) {
    float t;
    DPP_MIN(t, m, "quad_perm:[1,0,3,2]");   // xor 1
    DPP_MIN(m, t, "quad_perm:[2,3,0,1]");   // xor 2
    DPP_MIN(t, m, "row_half_mirror");        // merge quads within 8
    DPP_MIN(m, t, "row_mirror");             // merge 8s within 16
    int o;
    asm("v_permlanex16_b32 %0, %1, %2, %3"   // cross the two rows of 16
        : "=v"(o) : "v"(__float_as_int(m)), "s"(0x76543210), "s"(0xFEDCBA98));
    return vminf(m, __int_as_float(o));
}

// Build WMMA operands + per-batch segment bitmaps. One thread per (block, atom16).
// A layout (16x4 f32): lane m (0-15) vgpr v -> A[m][v]; lane 16+m vgpr v -> A[m][2+v]
// B layout (4x16 f32): lane n (0-15) vgpr v -> B[v][n]; lane 16+n vgpr v -> B[2+v][n]
__global__ void prep_kernel(const float* __restrict__ pos, const int* __restrict__ ft,
                            float* __restrict__ ws) {
    int t = blockIdx.x * blockDim.x + threadIdx.x;
    if (t >= NB * 16) return;
    int atom = t & 15;
    int g    = t >> 4;          // global block id
    int blk  = g & (LL - 1);    // block within batch
    float x, y, z;
    if (atom < AAT) {
        const float* p = pos + (g * AAT + atom) * 3;
        x = p[0]; y = p[1]; z = p[2];
    } else {
        // pad atom: far from all real atoms AND from every other block's pad
        x = 1.0e4f * (float)(blk + 1); y = 0.0f; z = 0.0f;
    }
    float x2 = x * x + y * y + z * z;
    float* Aarr = ws + WS_A;
    float* Barr = ws + WS_B;
    float* X2   = ws + WS_X2;
    X2[g * 16 + atom] = x2;
    int base = g * 32;
    // A-side: (x, y, z, 1)
    Aarr[(base + atom)      * 2 + 0] = x;
    Aarr[(base + atom)      * 2 + 1] = y;
    Aarr[(base + atom + 16) * 2 + 0] = z;
    Aarr[(base + atom + 16) * 2 + 1] = 1.0f;
    // B-side: (-2x, -2y, -2z, |x|^2)
    Barr[(base + atom)      * 2 + 0] = -2.0f * x;
    Barr[(base + atom)      * 2 + 1] = -2.0f * y;
    Barr[(base + atom + 16) * 2 + 0] = -2.0f * z;
    Barr[(base + atom + 16) * 2 + 1] = x2;

    // segment bitmap: 8 u32 words per batch (bit j = seg of block j)
    if (t < BB * 8) {
        int bb = t >> 3, w = t & 7;
        unsigned wd = 0u;
        for (int jj = 0; jj < 32; ++jj) {
            int s = ft[bb * LL + (w << 5) + jj];
            s = (s == 2) ? 1 : s;
            wd |= ((unsigned)(s & 1)) << jj;
        }
        ((unsigned*)(ws + WS_SEG))[t] = wd;
    }
}

// One wave per (batch, source block). d2 tile via V_WMMA_F32_16X16X4_F32,
// tile-min via fused DPP-min butterfly, then uniform 9-deep insertion KNN.
__global__ void __launch_bounds__(256) knn_kernel(const float* __restrict__ ws,
                                                  float* __restrict__ out) {
    int tid  = blockIdx.x * blockDim.x + threadIdx.x;
    int wave = tid >> 5;
    int lane = tid & 31;
    if (wave >= NB) return;
    int b = wave >> 8;
    int i = wave & (LL - 1);
    const float* Aarr = ws + WS_A;
    const float* Barr = ws + WS_B;
    const float* X2   = ws + WS_X2;
    const unsigned* segw = (const unsigned*)(ws + WS_SEG);

    v2f a = *(const v2f*)(Aarr + (wave * 32 + lane) * 2);
    // C init: c[r] = |x_i|^2 of row atom (r + 8*(lane>=16))
    int half8 = (lane >> 4) * 8;
    v8f c;
#pragma unroll
    for (int r = 0; r < 8; ++r) c[r] = X2[wave * 16 + half8 + r];

    unsigned si = (segw[b * 8 + (i >> 5)] >> (i & 31)) & 1u;

    float Da[KK], Db[KK]; int Ia[KK], Ib[KK];
#pragma unroll
    for (int k = 0; k < KK; ++k) { Da[k] = __builtin_inff(); Db[k] = __builtin_inff(); Ia[k] = 0; Ib[k] = 0; }

    const int jb = b * LL;
    const float* bp = Barr + ((size_t)jb * 32 + lane) * 2;  // +j*64 per block
    v2f bv = *(const v2f*)(bp);                              // j = 0

    for (int w = 0; w < 8; ++w) {
        unsigned word = segw[b * 8 + w];
#pragma unroll 8
        for (int jj = 0; jj < 32; ++jj) {
            int j = (w << 5) | jj;
            // prefetch next block's B operand (j=255 reads into X2 region: valid, unused)
            v2f bvn = *(const v2f*)(bp + (j + 1) * 64);
            // d2[m][n] = |xi_m|^2 + (-2 xi.xj + |xj_n|^2)
            v8f d = __builtin_amdgcn_wmma_f32_16x16x4_f32(false, a, false, bv,
                                                          (short)0, c, false, false);
            float t0 = vmin3f(d[0], d[1], d[2]);
            float t1 = vmin3f(d[3], d[4], d[5]);
            float t2 = vmin3f(d[6], d[7], t0);
            float m  = vminf(t1, t2);
            m = wave32_min(m);

            bool same = (((word >> jj) & 1u) == si);
            if (same) {
                if (j != i && m < Da[KK - 1]) {
                    Da[KK - 1] = m; Ia[KK - 1] = j;
#pragma unroll
                    for (int k = KK - 1; k > 0; --k)
                        if (Da[k] < Da[k - 1]) {
                            float td = Da[k]; Da[k] = Da[k - 1]; Da[k - 1] = td;
                            int   ti = Ia[k]; Ia[k] = Ia[k - 1]; Ia[k - 1] = ti;
                        }
                }
            } else {
                if (m < Db[KK - 1]) {
                    Db[KK - 1] = m; Ib[KK - 1] = j;
#pragma unroll
                    for (int k = KK - 1; k > 0; --k)
                        if (Db[k] < Db[k - 1]) {
                            float td = Db[k]; Db[k] = Db[k - 1]; Db[k - 1] = td;
                            int   ti = Ib[k]; Ib[k] = Ib[k - 1]; Ib[k - 1] = ti;
                        }
                }
            }
            bv = bvn;
        }
    }

    if (lane == 0) {
        float src = (float)wave;
#pragma unroll
        for (int k = 0; k < KK; ++k) {
            int e0 = wave * KK + k;            // intra edge slot
            out[OFS_EDGES + e0]      = src;
            out[OFS_EDGES + NE + e0] = (float)(jb + Ia[k]);
            int e1 = NEH + e0;                 // inter edge slot
            out[OFS_EDGES + e1]      = src;
            out[OFS_EDGES + NE + e1] = (float)(jb + Ib[k]);
        }
    }
}

// block_id, batch_id, edge_attr broadcast
__global__ void misc_kernel(const float* __restrict__ emb, float* __restrict__ out) {
    int t = blockIdx.x * blockDim.x + threadIdx.x;
    if (t < N_BLOCKID) out[t] = (float)(t / AAT);
    if (t < NB)        out[OFS_BATCH + t] = (float)(t >> 8);
    if (t < NE * ES) {
        int e = t >> 4; int col = t & 15;
        out[OFS_ATTR + t] = emb[((e >= NEH) ? ES : 0) + col];
    }
}

extern "C" void kernel_launch(void* const* d_in, const int* in_sizes, int n_in,
                              void* d_out, int out_size, void* d_ws, size_t ws_size,
                              hipStream_t stream) {
    const float* pos = (const float*)d_in[0];   // pos_heavyatom [B,L,15,3] f32
    const int*   ft  = (const int*)d_in[6];     // fragment_type [B,L] i32
    const float* emb = (const float*)d_in[7];   // edge_emb [2,16] f32
    float* out = (float*)d_out;
    float* ws  = (float*)d_ws;

    prep_kernel<<<(NB * 16 + 255) / 256, 256, 0, stream>>>(pos, ft, ws);
    knn_kernel<<<(NB * 32) / 256, 256, 0, stream>>>(ws, out);
    misc_kernel<<<(NE * ES + 255) / 256, 256, 0, stream>>>(emb, out);
}